// MaxPoolingAggregatorSp_36636071035050
// MI455X (gfx1250) — compile-verified
//
#include <hip/hip_runtime.h>
#include <hip/hip_bf16.h>
#include <stdint.h>

#define THREADS        256
#define WAVESZ         32
#define ROWS_PER_WAVE  64
#define WAVES_PER_BLK  (THREADS / WAVESZ)                 // 8
#define ROWS_PER_BLK   (WAVES_PER_BLK * ROWS_PER_WAVE)    // 512
#define DIM            64

typedef float vfloat2 __attribute__((ext_vector_type(2)));

// Non-returning device-scope float max atomic (CDNA5 global_atomic_max_num_f32,
// tracked with STOREcnt). Addr is a generic pointer to global memory.
__device__ __forceinline__ void atomic_max_f32(float* addr, float v) {
    asm volatile("global_atomic_max_num_f32 %0, %1, off scope:SCOPE_DEV"
                 :: "v"(addr), "v"(v)
                 : "memory");
}

__global__ void __launch_bounds__(THREADS)
seg_init_kernel(float* __restrict__ out, int n) {
    int i = blockIdx.x * THREADS + threadIdx.x;
    int stride = gridDim.x * THREADS;
    const float ninf = -__builtin_inff();
    for (; i < n; i += stride) {
        __builtin_nontemporal_store(ninf, out + i);
    }
}

__global__ void __launch_bounds__(THREADS)
seg_max_kernel(const float* __restrict__ x,
               const int* __restrict__ vid,
               float* __restrict__ out,
               int nrows) {
    __shared__ int s_ids[ROWS_PER_BLK];

    const int tid  = threadIdx.x;
    const int row0 = blockIdx.x * ROWS_PER_BLK;

    // ---- Stage this block's vertex ids global -> LDS with async DMA loads
    // (gfx1250 GLOBAL_LOAD_ASYNC_TO_LDS_B32, tracked by ASYNCcnt). Exactly two
    // full-EXEC iterations: 256 threads x 2 = 512 ids. Tail rows clamp the
    // global address in-bounds; the scan loop never consumes those slots.
    #pragma unroll
    for (int i = tid; i < ROWS_PER_BLK; i += THREADS) {
        int r  = row0 + i;
        int rc = (r < nrows) ? r : (nrows - 1);
        const int* gsrc  = vid + rc;
        unsigned lds_off = (unsigned)(size_t)(&s_ids[i]);
        asm volatile("global_load_async_to_lds_b32 %0, %1, off"
                     :: "v"(lds_off), "v"(gsrc)
                     : "memory");
    }
    asm volatile("s_wait_asynccnt 0" ::: "memory");
    __syncthreads();

    // ---- Sorted-segment scan: one wave32 per 64 contiguous rows, lane owns
    // feature pair [2*lane, 2*lane+1]. Running max in registers; flush with a
    // device-scope float-max atomic only when the segment id changes.
    const int wave = tid >> 5;
    const int lane = tid & (WAVESZ - 1);
    const int c0   = lane * 2;
    const int rbase = row0 + wave * ROWS_PER_WAVE;
    const int lbase = wave * ROWS_PER_WAVE;

    int   cur = -1;
    float m0  = -__builtin_inff();
    float m1  = -__builtin_inff();

    for (int j = 0; j < ROWS_PER_WAVE; ++j) {
        int r = rbase + j;
        if (r >= nrows) break;                 // wave-uniform exit
        int id = s_ids[lbase + j];             // ds_load, broadcast (uniform addr)
        if (id != cur) {                       // wave-uniform branch (sorted ids)
            if (cur >= 0) {
                float* o = out + (size_t)cur * DIM + c0;
                atomic_max_f32(o, m0);
                atomic_max_f32(o + 1, m1);
            }
            cur = id;
            m0 = -__builtin_inff();
            m1 = -__builtin_inff();
        }
        // Streaming read: x_sp (256 MB) is consumed exactly once -> NT hint,
        // 8B per lane = one coalesced 256B global_load_b64 per wave per row.
        vfloat2 v = __builtin_nontemporal_load(
            (const vfloat2*)(x + (size_t)r * DIM) + lane);
        m0 = fmaxf(m0, v.x);
        m1 = fmaxf(m1, v.y);
    }
    if (cur >= 0) {
        float* o = out + (size_t)cur * DIM + c0;
        atomic_max_f32(o, m0);
        atomic_max_f32(o + 1, m1);
    }
}

extern "C" void kernel_launch(void* const* d_in, const int* in_sizes, int n_in,
                              void* d_out, int out_size, void* d_ws, size_t ws_size,
                              hipStream_t stream) {
    const float* x   = (const float*)d_in[0];
    const int*   vid = (const int*)d_in[1];
    float*       out = (float*)d_out;
    const int nrows  = in_sizes[1];   // E = number of edge rows

    // 1) Initialize output to -inf (empty-segment semantics + atomic-max identity).
    {
        int n      = out_size;
        int blocks = (n + THREADS - 1) / THREADS;
        if (blocks > 4096) blocks = 4096;     // grid-stride fill
        seg_init_kernel<<<blocks, THREADS, 0, stream>>>(out, n);
    }

    // 2) Sorted segment-max scatter.
    {
        int blocks = (nrows + ROWS_PER_BLK - 1) / ROWS_PER_BLK;
        seg_max_kernel<<<blocks, THREADS, 0, stream>>>(x, vid, out, nrows);
    }
}